// ConvLatticeModule_25400436588640
// MI455X (gfx1250) — compile-verified
//
#include <hip/hip_runtime.h>

typedef float v2f __attribute__((ext_vector_type(2)));
typedef float v8f __attribute__((ext_vector_type(8)));

#define D   32    // val_dim
#define KNB 9     // filter extent
#define F   32    // nr_filters
#define KD  288   // KNB * D
#define NFRAG (KD / 4 * 2)   // 144 B-fragments (72 k-quads x 2 n-tiles)

// One wave computes a 16-vertex x 32-filter tile with v_wmma_f32_16x16x4_f32.
// A-fragments: direct b64 gather loads (layout matches 32-bit 16x4 A striping).
// B-fragments: weight pre-swizzled in LDS into per-fragment contiguous 256B
// blocks -> one conflict-free ds_load_b64 per fragment, immediate offsets only.
__launch_bounds__(256, 2)
__global__ void lattice_conv_wmma(const float* __restrict__ lattice,
                                  const int*   __restrict__ nbr,
                                  const float* __restrict__ weight,
                                  const float* __restrict__ bias,
                                  float*       __restrict__ out,
                                  int nVerts) {
  // wswz[frag][lane][2]: frag = (k>>2)*2 + ntile, lane = (k&2)*8 + n,
  // last dim = k&1.  One fragment = 64 floats = exactly one wave's B operand.
  __shared__ float wswz[KD * F];   // 36 KB
  __shared__ float blds[F];

  const int tid = threadIdx.x;
  // Cooperative swizzled stage: b128 source loads, 9 iters/thread.
  for (int q = tid; q < (KD * F) / 4; q += 256) {
    const int i = q * 4;                       // 16B-aligned source index
    const float4 w4 = *(const float4*)(weight + i);
    const int k   = i >> 5;                    // 0..287
    const int col = i & 31;                    // multiple of 4
    const int t   = col >> 4;                  // n-tile
    const int n0  = col & 15;
    const int kk  = k & 3;
    const int base = (((k >> 2) * 2 + t) << 6) + ((kk >> 1) << 5) + (kk & 1);
    wswz[base + (n0 + 0) * 2] = w4.x;
    wswz[base + (n0 + 1) * 2] = w4.y;
    wswz[base + (n0 + 2) * 2] = w4.z;
    wswz[base + (n0 + 3) * 2] = w4.w;
  }
  if (tid < F) blds[tid] = bias[tid];
  __syncthreads();

  const int lane = tid & 31;
  const int wave = tid >> 5;
  const int half = lane >> 4;          // 0: lanes 0-15, 1: lanes 16-31
  const int m    = lane & 15;          // A-row (vertex in tile) / B,D column

  const int tile  = blockIdx.x * 8 + wave;
  const int vbase = tile * 16;
  if (vbase >= nVerts) return;         // wave-uniform scalar branch

  v8f acc0 = {};                       // filters 0..15
  v8f acc1 = {};                       // filters 16..31

  int vrow = vbase + m;
  if (vrow >= nVerts) vrow = nVerts - 1;

  // All 9 gather indices up front: breaks the load->addr->load chain.
  int idxs[KNB];
#pragma unroll
  for (int nb = 0; nb < KNB; ++nb)
    idxs[nb] = __builtin_nontemporal_load(&nbr[vrow * KNB + nb]);

  const float* wfrag = &wswz[lane * 2];   // single DS base, imm offsets below

  // 2-deep software pipeline over neighbors (fully unrolled -> static regs).
  v2f abuf[2][8];
  {
    const float* rp = lattice + (size_t)idxs[0] * D + half * 2;
#pragma unroll
    for (int kc = 0; kc < 8; ++kc)
      abuf[0][kc] = *(const v2f*)(rp + kc * 4);
  }

#pragma unroll
  for (int nb = 0; nb < KNB; ++nb) {
    const int cur = nb & 1;
    if (nb + 1 < KNB) {                 // prefetch next row while we compute
      const float* rp = lattice + (size_t)idxs[nb + 1] * D + half * 2;
#pragma unroll
      for (int kc = 0; kc < 8; ++kc)
        abuf[cur ^ 1][kc] = *(const v2f*)(rp + kc * 4);
    }
#pragma unroll
    for (int kc = 0; kc < 8; ++kc) {
      const int fidx = (nb * 8 + kc) * 2;      // fragment pair index
      v2f b0 = *(const v2f*)(wfrag + (fidx + 0) * 64);  // ds_load_b64, imm ofs
      v2f b1 = *(const v2f*)(wfrag + (fidx + 1) * 64);
      acc0 = __builtin_amdgcn_wmma_f32_16x16x4_f32(
          false, abuf[cur][kc], false, b0, (short)0, acc0, false, false);
      acc1 = __builtin_amdgcn_wmma_f32_16x16x4_f32(
          false, abuf[cur][kc], false, b1, (short)0, acc1, false, false);
    }
  }

  // C/D layout: VGPR j, lanes 0-15 -> row vbase+j, lanes 16-31 -> vbase+8+j.
  const float bv0 = blds[m];
  const float bv1 = blds[16 + m];
  float* ob = out + (size_t)(vbase + half * 8) * F + m;
#pragma unroll
  for (int j = 0; j < 8; ++j) {
    if (vbase + half * 8 + j < nVerts) {
      __builtin_nontemporal_store(acc0[j] + bv0, ob + j * F);
      __builtin_nontemporal_store(acc1[j] + bv1, ob + j * F + 16);
    }
  }
}

extern "C" void kernel_launch(void* const* d_in, const int* in_sizes, int n_in,
                              void* d_out, int out_size, void* d_ws, size_t ws_size,
                              hipStream_t stream) {
  const float* lattice = (const float*)d_in[0];   // [N, D] f32
  const int*   nbr     = (const int*)  d_in[1];   // [N, K] i32
  const float* weight  = (const float*)d_in[2];   // [K*D, F] f32
  const float* bias    = (const float*)d_in[3];   // [F] f32
  float*       out     = (float*)d_out;           // [N, F] f32

  const int nVerts = in_sizes[0] / D;             // N = 1,000,000
  const int tiles  = (nVerts + 15) / 16;
  const int blocks = (tiles + 7) / 8;             // 8 wave-tiles per block
  lattice_conv_wmma<<<blocks, 256, 0, stream>>>(lattice, nbr, weight, bias,
                                                out, nVerts);
}